// DeepRare_87943750352924
// MI455X (gfx1250) — compile-verified
//
#include <hip/hip_runtime.h>
#include <cfloat>
#include <math.h>

// DeepRare saliency pipeline for MI455X (gfx1250, wave32).
// Memory-bound: 3 streaming passes over ~54MB of inputs (L2-resident after pass 1).
// Bilinear resize implemented as two f32 WMMA GEMMs (V_WMMA_F32_16X16X4_F32).

typedef __attribute__((ext_vector_type(2))) float v2f;
typedef __attribute__((ext_vector_type(8))) float v8f;

// ---------------- bilinear interpolation matrix (240 x Sp, cols >= S are zero) ----------------
__global__ void buildR_kernel(float* __restrict__ R, int S, int Sp) {
  int i = blockIdx.x * blockDim.x + threadIdx.x;
  if (i >= 240 * Sp) return;
  int row = i / Sp, k = i % Sp;
  float wgt = 0.f;
  float c = (row + 0.5f) * ((float)S / 240.f) - 0.5f;
  float kf = floorf(c);
  int k0 = (int)kf;
  float f = c - kf;
  int ka = min(max(k0, 0), S - 1);
  int kb = min(max(k0 + 1, 0), S - 1);
  if (k == ka) wgt += 1.f - f;
  if (k == kb) wgt += f;
  R[i] = wgt;
}

// ---------------- pass 1+2 (fused): per-channel min/max then histograms ----------------
// one block (256 thr) per channel. counts: [hist6 | lutAll6 | lutInterior6]
__global__ void hist_kernel(const float* __restrict__ in, float* __restrict__ chstat,
                            int* __restrict__ counts, int chanBase, int H, int W) {
  __shared__ float red[256];
  __shared__ float s_mn, s_mx;
  __shared__ int s_cnt[18];
  const int tid = threadIdx.x;
  const int c = blockIdx.x;
  const int HW = H * W;
  const float* src = in + (size_t)c * HW;
  float mn = FLT_MAX, mx = -FLT_MAX;
  for (int i = tid; i < HW; i += 256) {
    __builtin_prefetch(src + i + 4096, 0, 0);  // global_prefetch_b8
    int y = i / W, x = i - y * W;
    float v = src[i];
    if (y == 0 || y == H - 1 || x == 0 || x == W - 1) v = 0.f;
    mn = fminf(mn, v); mx = fmaxf(mx, v);
  }
  red[tid] = mn; __syncthreads();
  for (int s = 128; s > 0; s >>= 1) { if (tid < s) red[tid] = fminf(red[tid], red[tid + s]); __syncthreads(); }
  if (tid == 0) s_mn = red[0];
  __syncthreads();
  red[tid] = mx; __syncthreads();
  for (int s = 128; s > 0; s >>= 1) { if (tid < s) red[tid] = fmaxf(red[tid], red[tid + s]); __syncthreads(); }
  if (tid == 0) s_mx = red[0];
  if (tid < 18) s_cnt[tid] = 0;
  __syncthreads();
  mn = s_mn; mx = s_mx;
  float d = mx - mn;
  float scale = (d == 0.f) ? 0.f : (256.f / d);
  for (int i = tid; i < HW; i += 256) {
    int y = i / W, x = i - y * W;
    float v = src[i];
    bool border = (y == 0 || y == H - 1 || x == 0 || x == W - 1);
    if (border) v = 0.f;
    float nrm = (v - mn) * scale;                    // 0..256 (zeros if degenerate)
    int b  = (int)(nrm * (6.f / 256.f)); b  = min(max(b, 0), 5);
    int li = (int)(nrm * 6.f - 1.f);     li = min(max(li, 0), 5);
    atomicAdd(&s_cnt[b], 1);
    atomicAdd(&s_cnt[6 + li], 1);
    if (!border) atomicAdd(&s_cnt[12 + li], 1);
  }
  __syncthreads();
  if (tid == 0) { chstat[(size_t)(chanBase + c) * 2] = mn; chstat[(size_t)(chanBase + c) * 2 + 1] = mx; }
  if (tid < 18) counts[(size_t)(chanBase + c) * 18 + tid] = s_cnt[tid];
}

// ---------------- per-channel analytic LUT (rarity + both ponder stages) ----------------
// clut per channel (stride 10): [mn, scale, lut0..lut5, bordval, pad]
__global__ void chan_lut_kernel(const int* __restrict__ counts, const float* __restrict__ chstat,
                                float* __restrict__ clut, int chanBase, int C, int HW) {
  int c = blockIdx.x * blockDim.x + threadIdx.x;
  if (c >= C) return;
  int gc = chanBase + c;
  const int* h = counts + (size_t)gc * 18;
  float tot = (float)HW;
  float nh[6], rv[6];
  int la[6], lint[6];
  for (int j = 0; j < 6; ++j) {
    la[j] = h[6 + j]; lint[j] = h[12 + j];
    nh[j] = -logf((float)h[j] / tot + 1e-4f);
  }
  float dmin = FLT_MAX, dmax = -FLT_MAX;
  for (int j = 0; j < 6; ++j) if (la[j] > 0) { dmin = fminf(dmin, nh[j]); dmax = fmaxf(dmax, nh[j]); }
  float dd = dmax - dmin;
  if (dd == 0.f) {
    for (int j = 0; j < 6; ++j) rv[j] = 0.f;
  } else {
    float dn[6]; float maxdn = -FLT_MAX, meandn = 0.f;
    for (int j = 0; j < 6; ++j) {
      dn[j] = (nh[j] - dmin) / dd;
      if (la[j] > 0) maxdn = fmaxf(maxdn, dn[j]);
      meandn += (float)la[j] * dn[j];
    }
    meandn /= tot;
    float wq = maxdn - meandn; wq *= wq;
    for (int j = 0; j < 6; ++j) rv[j] = dn[j] * wq;   // r-value per lut bin
  }
  float lut[6]; float bordv = 0.f;
  if (c == 0) {  // full-map ponder of r[0]
    float m0 = FLT_MAX, M0 = -FLT_MAX, mean0 = 0.f;
    for (int j = 0; j < 6; ++j) { if (la[j] > 0) { m0 = fminf(m0, rv[j]); M0 = fmaxf(M0, rv[j]); } mean0 += (float)la[j] * rv[j]; }
    mean0 /= tot;
    float D = M0 - m0;
    if (D == 0.f) { for (int j = 0; j < 6; ++j) lut[j] = 0.f; }
    else { float w2 = M0 - mean0; w2 *= w2; for (int j = 0; j < 6; ++j) lut[j] = (rv[j] - m0) / D * w2; }
  } else {       // border-masked ponder of r[c]
    int ni = 0; for (int j = 0; j < 6; ++j) ni += lint[j];
    int nb = HW - ni;
    float mmn = FLT_MAX, mmx = -FLT_MAX, mean = 0.f;
    if (nb > 0) { mmn = 0.f; mmx = 0.f; }
    for (int j = 0; j < 6; ++j) { if (lint[j] > 0) { mmn = fminf(mmn, rv[j]); mmx = fmaxf(mmx, rv[j]); } mean += (float)lint[j] * rv[j]; }
    mean /= tot;  // border pixels contribute 0
    float D = mmx - mmn;
    if (D == 0.f) { for (int j = 0; j < 6; ++j) lut[j] = 0.f; bordv = 0.f; }
    else { float w2 = mmx - mean; w2 *= w2; for (int j = 0; j < 6; ++j) lut[j] = (rv[j] - mmn) / D * w2; bordv = (0.f - mmn) / D * w2; }
  }
  float mn = chstat[(size_t)gc * 2], mx = chstat[(size_t)gc * 2 + 1];
  float d = mx - mn;
  float scale = (d == 0.f) ? 0.f : (256.f / d);
  float* o = clut + (size_t)gc * 10;
  o[0] = mn; o[1] = scale;
  for (int j = 0; j < 6; ++j) o[2 + j] = lut[j];
  o[8] = bordv; o[9] = 0.f;
}

// ---------------- pass 3: per-pixel gather-accumulate over channels -> padded p map ----------------
__global__ void accumulate_kernel(const float* __restrict__ in, const float* __restrict__ clut,
                                  float* __restrict__ pPad, int chanBase, int C, int H, int W, int Sp) {
  extern __shared__ float sStats[];  // C*10 floats
  int total = C * 10;
  for (int i = threadIdx.x; i < total; i += blockDim.x) sStats[i] = clut[(size_t)chanBase * 10 + i];
  __syncthreads();
  int idx = blockIdx.x * blockDim.x + threadIdx.x;
  if (idx >= Sp * Sp) return;
  int y = idx / Sp, x = idx - y * Sp;
  if (y >= H || x >= W) { pPad[idx] = 0.f; return; }  // zero matrix padding
  bool border = (y == 0) || (y == H - 1) || (x == 0) || (x == W - 1);
  int HW = H * W;
  int pix = y * W + x;
  float acc = 0.f;
  for (int c = 0; c < C; ++c) {
    const float* s = sStats + c * 10;
    float v = border ? 0.f : in[(size_t)c * HW + pix];
    float nrm = (v - s[0]) * s[1];
    int li = (int)(nrm * 6.f - 1.f);
    li = min(max(li, 0), 5);
    acc += (c == 0 || !border) ? s[2 + li] : s[8];
  }
  pPad[idx] = acc;
}

// ---------------- reductions ----------------
__global__ void pminmax_kernel(const float* __restrict__ p, float* __restrict__ out2, int H, int W, int Sp) {
  __shared__ float red[1024];
  int tid = threadIdx.x;
  float mn = FLT_MAX, mx = -FLT_MAX;
  int HW = H * W;
  for (int i = tid; i < HW; i += 1024) {
    int y = i / W, x = i - y * W;
    float v = p[(size_t)y * Sp + x];
    mn = fminf(mn, v); mx = fmaxf(mx, v);
  }
  red[tid] = mn; __syncthreads();
  for (int s = 512; s > 0; s >>= 1) { if (tid < s) red[tid] = fminf(red[tid], red[tid + s]); __syncthreads(); }
  if (tid == 0) out2[0] = red[0];
  __syncthreads();
  red[tid] = mx; __syncthreads();
  for (int s = 512; s > 0; s >>= 1) { if (tid < s) red[tid] = fmaxf(red[tid], red[tid + s]); __syncthreads(); }
  if (tid == 0) out2[1] = red[0];
}

__global__ void pnorm_kernel(float* __restrict__ pPad, const float* __restrict__ st, int H, int W, int Sp) {
  int i = blockIdx.x * blockDim.x + threadIdx.x;
  if (i >= H * W) return;
  int y = i / W, x = i - y * W;
  float mn = st[0], mx = st[1];
  float d = mx - mn;
  float v = pPad[(size_t)y * Sp + x];
  pPad[(size_t)y * Sp + x] = (d == 0.f) ? 0.f : ((v - mn) / d);
}

__global__ void mapstats_kernel(const float* __restrict__ m, float* __restrict__ out4) {
  __shared__ float red[1024];
  int tid = threadIdx.x;
  float mn = FLT_MAX, mx = -FLT_MAX, sm = 0.f;
  for (int i = tid; i < 57600; i += 1024) { float v = m[i]; mn = fminf(mn, v); mx = fmaxf(mx, v); sm += v; }
  red[tid] = mn; __syncthreads();
  for (int s = 512; s > 0; s >>= 1) { if (tid < s) red[tid] = fminf(red[tid], red[tid + s]); __syncthreads(); }
  if (tid == 0) out4[0] = red[0];
  __syncthreads();
  red[tid] = mx; __syncthreads();
  for (int s = 512; s > 0; s >>= 1) { if (tid < s) red[tid] = fmaxf(red[tid], red[tid + s]); __syncthreads(); }
  if (tid == 0) out4[1] = red[0];
  __syncthreads();
  red[tid] = sm; __syncthreads();
  for (int s = 512; s > 0; s >>= 1) { if (tid < s) red[tid] += red[tid + s]; __syncthreads(); }
  if (tid == 0) out4[2] = red[0];
}

// ---------------- WMMA f32 GEMM: D(M x N) = A(M x K) * B ----------------
// BT=false: B is K x N row-major (ldb = N). BT=true: B(k,n) = Bsrc[n*ldb + k].
// One wave per 16x16 tile; V_WMMA_F32_16X16X4_F32, K stepped by 4.
// f32 A 16x4 layout: lanes 0-15 -> K={0,1} in v0/v1; lanes 16-31 -> K={2,3}. B mirrored.
template <bool BT>
__global__ void wmma_gemm_kernel(const float* __restrict__ A, const float* __restrict__ B,
                                 float* __restrict__ D, int N, int K, int lda, int ldb, int ldd) {
  int tilesN = N >> 4;
  int tn = blockIdx.x % tilesN;
  int tm = blockIdx.x / tilesN;
  int lane = threadIdx.x;
  int l15 = lane & 15;
  int hi = lane >> 4;
  int m0 = tm * 16, n0 = tn * 16;
  v8f acc = {0.f, 0.f, 0.f, 0.f, 0.f, 0.f, 0.f, 0.f};
  const float* arow = A + (size_t)(m0 + l15) * lda;
  const float* brow = BT ? (B + (size_t)(n0 + l15) * ldb) : nullptr;
  for (int k = 0; k < K; k += 4) {
    int ka = k + 2 * hi;
    v2f a; a.x = arow[ka]; a.y = arow[ka + 1];
    v2f b;
    if (BT) { b.x = brow[ka]; b.y = brow[ka + 1]; }
    else    { b.x = B[(size_t)ka * ldb + n0 + l15]; b.y = B[(size_t)(ka + 1) * ldb + n0 + l15]; }
    acc = __builtin_amdgcn_wmma_f32_16x16x4_f32(false, a, false, b, (short)0, acc, false, false);
  }
  for (int r = 0; r < 8; ++r)
    D[(size_t)(m0 + r + 8 * hi) * ldd + n0 + l15] = acc[r];
}

// ---------------- group fuse + final ----------------
__global__ void fuse_kernel(const float* __restrict__ maps, const float* __restrict__ mst,
                            int cnt, float* __restrict__ fused) {
  int i = blockIdx.x * blockDim.x + threadIdx.x;
  if (i >= 57600) return;
  float acc = 0.f;
  for (int m = 0; m < cnt; ++m) {
    float mn = mst[m * 4], mx = mst[m * 4 + 1];
    float mean = mst[m * 4 + 2] * (1.f / 57600.f);
    float d = mx - mn;
    if (d != 0.f) {
      float w = mx - mean; w *= w;
      acc += (maps[(size_t)m * 57600 + i] - mn) / d * w;
    }
  }
  fused[i] = acc;
}

__global__ void final_kernel(const float* __restrict__ fused, const float* __restrict__ gstat,
                             float* __restrict__ out) {
  int i = blockIdx.x * blockDim.x + threadIdx.x;
  if (i >= 57600) return;
  float s = 0.f;
  for (int g = 0; g < 5; ++g) {
    float mn = gstat[g * 2], mx = gstat[g * 2 + 1];
    float d = mx - mn;
    float v = 0.f;
    if (d != 0.f) v = (fused[(size_t)g * 57600 + i] - mn) / d * 256.f;
    out[57600 + (size_t)i * 5 + g] = v;  // groups: (240,240,5)
    s += v;
  }
  out[i] = s;  // groups.sum(-1)
}

// ---------------- host launcher ----------------
extern "C" void kernel_launch(void* const* d_in, const int* in_sizes, int n_in,
                              void* d_out, int out_size, void* d_ws, size_t ws_size,
                              hipStream_t stream) {
  (void)in_sizes; (void)n_in; (void)out_size; (void)ws_size;
  static const int Cs[13]  = {64, 64, 128, 128, 256, 256, 256, 512, 512, 512, 512, 512, 512};
  static const int SIx[13] = {0, 0, 1, 1, 2, 2, 2, 3, 3, 3, 4, 4, 4};
  static const int S_of[5]  = {224, 112, 56, 28, 14};
  static const int Sp_of[5] = {224, 112, 64, 32, 16};   // padded to 16 (WMMA tiles)
  static const int Gfirst[5] = {0, 2, 4, 7, 10};
  static const int Gcnt[5]   = {2, 2, 3, 3, 3};

  int chanBase[13]; int cb = 0;
  for (int t = 0; t < 13; ++t) { chanBase[t] = cb; cb += Cs[t]; }
  const int totCh = cb;  // 4224

  // workspace carve (~7 MB of floats)
  float* w = (float*)d_ws;
  size_t off = 0;
  auto alloc = [&](size_t n) { float* p = w + off; off += n; return p; };
  float* chstat = alloc((size_t)totCh * 2);
  float* clut   = alloc((size_t)totCh * 10);
  int*   counts = (int*)alloc((size_t)totCh * 18);
  float* Rm[5];
  for (int s = 0; s < 5; ++s) Rm[s] = alloc((size_t)240 * Sp_of[s]);
  float* pPad[13]; float* Tt[13];
  for (int t = 0; t < 13; ++t) { int Sp = Sp_of[SIx[t]]; pPad[t] = alloc((size_t)Sp * Sp); }
  for (int t = 0; t < 13; ++t) { int Sp = Sp_of[SIx[t]]; Tt[t]  = alloc((size_t)240 * Sp); }
  float* res   = alloc((size_t)13 * 57600);
  float* pst   = alloc((size_t)13 * 2);
  float* mst   = alloc((size_t)13 * 4);
  float* fused = alloc((size_t)5 * 57600);
  float* gst   = alloc((size_t)5 * 2);

  // 1) interpolation matrices
  for (int s = 0; s < 5; ++s) {
    int n = 240 * Sp_of[s];
    buildR_kernel<<<(n + 255) / 256, 256, 0, stream>>>(Rm[s], S_of[s], Sp_of[s]);
  }
  // 2) per-channel min/max + histograms
  for (int t = 0; t < 13; ++t) {
    int S = S_of[SIx[t]];
    hist_kernel<<<Cs[t], 256, 0, stream>>>((const float*)d_in[t], chstat, counts, chanBase[t], S, S);
  }
  // 3) per-channel analytic LUTs
  for (int t = 0; t < 13; ++t) {
    int S = S_of[SIx[t]];
    chan_lut_kernel<<<(Cs[t] + 127) / 128, 128, 0, stream>>>(counts, chstat, clut, chanBase[t], Cs[t], S * S);
  }
  // 4) gather-accumulate -> padded p map; 5) min/max; 6) normalize
  for (int t = 0; t < 13; ++t) {
    int S = S_of[SIx[t]], Sp = Sp_of[SIx[t]];
    int n = Sp * Sp;
    accumulate_kernel<<<(n + 255) / 256, 256, (size_t)Cs[t] * 10 * sizeof(float), stream>>>(
        (const float*)d_in[t], clut, pPad[t], chanBase[t], Cs[t], S, S, Sp);
    pminmax_kernel<<<1, 1024, 0, stream>>>(pPad[t], pst + t * 2, S, S, Sp);
    pnorm_kernel<<<(S * S + 255) / 256, 256, 0, stream>>>(pPad[t], pst + t * 2, S, S, Sp);
  }
  // 7) resize240 = Rh * P * Rw^T via WMMA GEMMs; 8) per-map stats
  for (int t = 0; t < 13; ++t) {
    int si = SIx[t], Sp = Sp_of[si];
    wmma_gemm_kernel<false><<<15 * (Sp / 16), 32, 0, stream>>>(Rm[si], pPad[t], Tt[t], Sp, Sp, Sp, Sp, Sp);
    wmma_gemm_kernel<true><<<15 * 15, 32, 0, stream>>>(Tt[t], Rm[si], res + (size_t)t * 57600, 240, Sp, Sp, Sp, 240);
    mapstats_kernel<<<1, 1024, 0, stream>>>(res + (size_t)t * 57600, mst + t * 4);
  }
  // 9) group fuse + stats
  for (int g = 0; g < 5; ++g) {
    fuse_kernel<<<(57600 + 255) / 256, 256, 0, stream>>>(
        res + (size_t)Gfirst[g] * 57600, mst + Gfirst[g] * 4, Gcnt[g], fused + (size_t)g * 57600);
    pminmax_kernel<<<1, 1024, 0, stream>>>(fused + (size_t)g * 57600, gst + g * 2, 240, 240, 240);
  }
  // 10) final normalize + output layout
  final_kernel<<<(57600 + 255) / 256, 256, 0, stream>>>(fused, gst, (float*)d_out);
}